// SAModule_37168646979943
// MI455X (gfx1250) — compile-verified
//
#include <hip/hip_runtime.h>
#include <math.h>

// Problem constants (match reference)
#define B_     4
#define N_PER_ 4096
#define F_     64
#define M_PER_ 2048
#define K_     32
#define NQ_    (B_ * M_PER_)   // 8192 sampled queries total
#define R2_    0.04f           // R=0.2

typedef __attribute__((ext_vector_type(2))) float    v2f;
typedef __attribute__((ext_vector_type(8))) float    v8f;
typedef __attribute__((ext_vector_type(4))) unsigned v4u;
typedef __attribute__((ext_vector_type(8))) int      v8i;
typedef __attribute__((ext_vector_type(4))) int      v4i;

__device__ __forceinline__ v8f wmma4(v2f a, v2f b, v8f c) {
  // D(16x16,f32) = A(16x4,f32) * B(4x16,f32) + C   — fp32 matrix pipe
  return __builtin_amdgcn_wmma_f32_16x16x4_f32(false, a, false, b, (short)0, c, false, false);
}

// ---------------------------------------------------------------------------
// Phase 0: zero-pad W1 (67x64) to 68x64 so the WMMA K-loop needs no predicates.
// ---------------------------------------------------------------------------
__global__ void pad_w1_kernel(const float* __restrict__ W1, float* __restrict__ W1p)
{
  const int i = blockIdx.x * blockDim.x + threadIdx.x;
  if (i < 68 * 64) W1p[i] = (i < 67 * 64) ? W1[i] : 0.0f;
}

// ---------------------------------------------------------------------------
// Phase 1: farthest point sampling. One workgroup per cloud; points + min-dists
// in registers; LDS tree reduction with first-max (lowest index) tie-break.
// ---------------------------------------------------------------------------
__global__ void __launch_bounds__(512)
fps_kernel(const float* __restrict__ pos, float* __restrict__ qpos,
           float* __restrict__ out_pos, int* __restrict__ out_batch)
{
  const int b = blockIdx.x, t = threadIdx.x;
  const float* P = pos + (size_t)b * N_PER_ * 3;

  float px[8], py[8], pz[8], mind[8];
  const int base = t * 8;
#pragma unroll
  for (int i = 0; i < 8; ++i) {
    px[i] = P[(base + i) * 3 + 0];
    py[i] = P[(base + i) * 3 + 1];
    pz[i] = P[(base + i) * 3 + 2];
  }

  __shared__ float bc[3];
  __shared__ float sval[512];
  __shared__ int   sidx[512];

  if (t == 0) { bc[0] = P[0]; bc[1] = P[1]; bc[2] = P[2]; }
  __syncthreads();
  float cx = bc[0], cy = bc[1], cz = bc[2];
#pragma unroll
  for (int i = 0; i < 8; ++i) {
    float dx = px[i] - cx, dy = py[i] - cy, dz = pz[i] - cz;
    mind[i] = dx * dx + dy * dy + dz * dz;
  }
  if (t == 0) {
    const int o = b * M_PER_;
    qpos[o * 3 + 0] = cx; qpos[o * 3 + 1] = cy; qpos[o * 3 + 2] = cz;
    out_pos[o * 3 + 0] = cx; out_pos[o * 3 + 1] = cy; out_pos[o * 3 + 2] = cz;
  }

  for (int s = 1; s < M_PER_; ++s) {
    float bv = mind[0]; int bi = base;
#pragma unroll
    for (int i = 1; i < 8; ++i)
      if (mind[i] > bv) { bv = mind[i]; bi = base + i; }
    sval[t] = bv; sidx[t] = bi;
    __syncthreads();
    for (int off = 256; off > 0; off >>= 1) {
      if (t < off) {
        float v2 = sval[t + off]; int i2 = sidx[t + off];
        float v1 = sval[t];       int i1 = sidx[t];
        if (v2 > v1 || (v2 == v1 && i2 < i1)) { sval[t] = v2; sidx[t] = i2; }
      }
      __syncthreads();
    }
    const int g = sidx[0];
    if ((g >> 3) == t) { const int l = g & 7; bc[0] = px[l]; bc[1] = py[l]; bc[2] = pz[l]; }
    __syncthreads();
    cx = bc[0]; cy = bc[1]; cz = bc[2];
    if (t == 0) {
      const int o = b * M_PER_ + s;
      qpos[o * 3 + 0] = cx; qpos[o * 3 + 1] = cy; qpos[o * 3 + 2] = cz;
      out_pos[o * 3 + 0] = cx; out_pos[o * 3 + 1] = cy; out_pos[o * 3 + 2] = cz;
    }
#pragma unroll
    for (int i = 0; i < 8; ++i) {
      float dx = px[i] - cx, dy = py[i] - cy, dz = pz[i] - cz;
      float d = dx * dx + dy * dy + dz * dz;
      mind[i] = fminf(mind[i], d);
    }
    __syncthreads();
  }

  for (int i = t; i < M_PER_; i += 512) out_batch[b * M_PER_ + i] = b;
}

// ---------------------------------------------------------------------------
// Phase 2: k=32 nearest neighbors per query. Pos tiles are DMA'd into LDS by
// the Tensor Data Mover (one tensor_load_to_lds per 12KB tile, wave 0 issues,
// s_wait_tensorcnt + barrier publishes). Sorted top-32 lists live in LDS.
// ---------------------------------------------------------------------------
#define KNN_T 128
#define TILE_ 1024

#if __has_builtin(__builtin_amdgcn_tensor_load_to_lds) && __has_builtin(__builtin_amdgcn_s_wait_tensorcnt)
#define USE_TDM_ 1
#else
#define USE_TDM_ 0
#endif

__global__ void __launch_bounds__(KNN_T)
knn_kernel(const float* __restrict__ pos, const float* __restrict__ qpos,
           int* __restrict__ nidx, unsigned* __restrict__ vmask)
{
  const int tid = threadIdx.x;
  const int q = blockIdx.x * KNN_T + tid;       // [0, 8192)
  const int b = q / M_PER_;                     // block-uniform (2048 % 128 == 0)
  const float* P = pos + (size_t)b * N_PER_ * 3;

  __shared__ float pt[TILE_ * 3];               // interleaved xyz tile (12 KB)
  __shared__ float dl[32 * KNN_T];
  __shared__ int   il[32 * KNN_T];

  const float qx = qpos[q * 3 + 0], qy = qpos[q * 3 + 1], qz = qpos[q * 3 + 2];
  for (int j = 0; j < 32; ++j) { dl[j * KNN_T + tid] = 3.402823466e38f; il[j * KNN_T + tid] = 0; }

  for (int t0 = 0; t0 < N_PER_; t0 += TILE_) {
    __syncthreads();                            // previous tile fully consumed
#if USE_TDM_
    if (tid < 32) {                             // wave 0 drives the TDM
      const unsigned lds_off = (unsigned)(size_t)(void*)pt;  // LDS offset = low 32 bits
      const unsigned long long ga = (unsigned long long)(const void*)(P + (size_t)t0 * 3);
      // D# group0: count=1 | lds_addr | global_addr[56:0] | type=2
      v4u g0 = { 1u, lds_off, (unsigned)ga,
                 (unsigned)((ga >> 32) & 0x01FFFFFFu) | 0x80000000u };
      // D# group1: data_size=4B; 2D tile: dim0=3072 elems, dim1=1 row
      const unsigned td0 = TILE_ * 3;           // 3072 elements
      v8i g1;
      g1[0] = (int)(2u << 16);                  // workgroup_mask=0, data_size=2 (4B)
      g1[1] = (int)((td0 & 0xFFFFu) << 16);     // tensor_dim0[15:0] in [31:16]
      g1[2] = (int)((td0 >> 16) | (1u << 16));  // tensor_dim0[31:16] | tensor_dim1=1
      g1[3] = (int)((td0 & 0xFFFFu) << 16);     // tensor_dim1[31:16]=0 | tile_dim0
      g1[4] = 1;                                // tile_dim1=1, tile_dim2=0
      g1[5] = (int)td0;                         // tensor_dim0_stride[31:0]
      g1[6] = 0;                                // stride0[47:32] | stride1[15:0]
      g1[7] = 0;
      v4i gz4 = { 0, 0, 0, 0 };                 // groups 2/3 unused (<=2D)
      v8i gz8 = { 0, 0, 0, 0, 0, 0, 0, 0 };     // extra group (6-arg toolchain form)
      __builtin_amdgcn_tensor_load_to_lds(g0, g1, gz4, gz4, gz8, 0);
      __builtin_amdgcn_s_wait_tensorcnt(0);
    }
#else
    for (int i = tid; i < TILE_ * 3; i += KNN_T) pt[i] = P[(size_t)t0 * 3 + i];
#endif
    __syncthreads();                            // publish tile to all waves
    for (int i = 0; i < TILE_; ++i) {           // uniform i -> LDS broadcast reads
      const float dx = pt[i * 3 + 0] - qx;
      const float dy = pt[i * 3 + 1] - qy;
      const float dz = pt[i * 3 + 2] - qz;
      const float d = dx * dx + dy * dy + dz * dz;
      if (d < dl[31 * KNN_T + tid]) {           // strict < : stability on ties
        int j = 31;
        while (j > 0 && dl[(j - 1) * KNN_T + tid] > d) {
          dl[j * KNN_T + tid] = dl[(j - 1) * KNN_T + tid];
          il[j * KNN_T + tid] = il[(j - 1) * KNN_T + tid];
          --j;
        }
        dl[j * KNN_T + tid] = d;
        il[j * KNN_T + tid] = b * N_PER_ + t0 + i;
      }
    }
  }

  unsigned m = 0;
  for (int j = 0; j < 32; ++j) {
    nidx[q * 32 + j] = il[j * KNN_T + tid];
    if (dl[j * KNN_T + tid] <= R2_) m |= (1u << j);
  }
  vmask[q] = m;
}

// ---------------------------------------------------------------------------
// Phase 3+4: PointNetConv MLP via V_WMMA_F32_16X16X4_F32.
// Loop order n-tile -> k-step -> {rowtile0,rowtile1}: each weight fragment
// feeds two WMMAs (two independent accumulator chains), no predicates, and the
// fully-unrolled k-loop lets the compiler pipeline global weight loads.
// ---------------------------------------------------------------------------
template<int KSTEPS, int FO, int SA, int SO>
__device__ __forceinline__ void mlp_layer(const float* Abuf,
                                          const float* __restrict__ W,
                                          const float* __restrict__ bias,
                                          float* Obuf, int lane)
{
  const int lmod = lane & 15, lhi = lane >> 4;
  const float* Ar0 = Abuf + lmod * SA;          // rows 0..15
  const float* Ar1 = Abuf + (lmod + 16) * SA;   // rows 16..31
  for (int n0 = 0; n0 < FO; n0 += 16) {
    const int n = n0 + lmod;
    const float bv = bias[n];
    v8f acc0 = {bv, bv, bv, bv, bv, bv, bv, bv};
    v8f acc1 = acc0;
    const float* Wp = W + (size_t)(2 * lhi) * FO + n;   // row k = 2*lhi
#pragma unroll
    for (int ks = 0; ks < KSTEPS; ++ks) {
      const int k = ks * 4 + 2 * lhi;
      v2f bf;
      bf.x = Wp[0];                             // W[k][n]
      bf.y = Wp[FO];                            // W[k+1][n]
      Wp += 4 * FO;
      v2f a0 = *(const v2f*)(Ar0 + k);
      v2f a1 = *(const v2f*)(Ar1 + k);
      acc0 = wmma4(a0, bf, acc0);
      acc1 = wmma4(a1, bf, acc1);
    }
#pragma unroll
    for (int i = 0; i < 8; ++i) {               // C/D: VGPR i -> M = i + 8*lhi
      const int row = i + 8 * lhi;
      const float v0 = acc0[i], v1 = acc1[i];
      Obuf[row * SO + n]        = v0 > 0.0f ? v0 : 0.0f;   // ReLU
      Obuf[(row + 16) * SO + n] = v1 > 0.0f ? v1 : 0.0f;
    }
  }
}

__global__ void __launch_bounds__(32)
mlp_kernel(const float* __restrict__ x, const float* __restrict__ pos,
           const float* __restrict__ qpos, const int* __restrict__ nidx,
           const unsigned* __restrict__ vmask,
           const float* __restrict__ W1p, const float* __restrict__ b1,
           const float* __restrict__ W2, const float* __restrict__ b2,
           const float* __restrict__ W3, const float* __restrict__ b3,
           const float* __restrict__ Wg, const float* __restrict__ bg,
           float* __restrict__ out_x)
{
  __shared__ float bufA[32 * 72];    // 32 x 68 activations (stride 72, even)
  __shared__ float bufB[32 * 132];   // up to 32 x 128 (stride 132, even)

  const int qi = blockIdx.x;
  const int lane = threadIdx.x;

  // Gather: lane L stages neighbor L's row [x(64) | rel(3) | 0]
  const int nj = nidx[qi * 32 + lane];
  const float* xr = x + (size_t)nj * F_;
  float* Ar = bufA + lane * 72;
#pragma unroll
  for (int c = 0; c < 64; c += 4) {
    float4 v = *(const float4*)(xr + c);
    Ar[c + 0] = v.x; Ar[c + 1] = v.y; Ar[c + 2] = v.z; Ar[c + 3] = v.w;
  }
  const float qx = qpos[qi * 3 + 0], qy = qpos[qi * 3 + 1], qz = qpos[qi * 3 + 2];
  Ar[64] = pos[nj * 3 + 0] - qx;
  Ar[65] = pos[nj * 3 + 1] - qy;
  Ar[66] = pos[nj * 3 + 2] - qz;
  Ar[67] = 0.0f;
  __syncthreads();

  mlp_layer<17,  64,  72, 132>(bufA, W1p, b1, bufB, lane);
  __syncthreads();
  mlp_layer<16,  64, 132,  72>(bufB, W2,  b2, bufA, lane);
  __syncthreads();
  mlp_layer<16, 128,  72, 132>(bufA, W3,  b3, bufB, lane);
  __syncthreads();

  // Masked max over 32 neighbors (vm wave-uniform -> uniform branch)
  const unsigned vm = vmask[qi];
  float aggv[4];
#pragma unroll
  for (int p = 0; p < 4; ++p) {
    const int c = lane + 32 * p;
    float mx = -3.402823466e38f;
    for (int r = 0; r < 32; ++r)
      if ((vm >> r) & 1u) { const float v = bufB[r * 132 + c]; mx = v > mx ? v : mx; }
    aggv[p] = mx;
  }
  __syncthreads();
#pragma unroll
  for (int p = 0; p < 4; ++p) bufA[lane + 32 * p] = aggv[p];
  __syncthreads();

  // Global linear: out = agg @ Wg + bg (4 output cols per lane)
#pragma unroll
  for (int p = 0; p < 4; ++p) {
    const int c = lane + 32 * p;
    float s = bg[c];
    for (int k = 0; k < 128; ++k) s += bufA[k] * Wg[k * 128 + c];
    out_x[(size_t)qi * 128 + c] = s;
  }
}

// ---------------------------------------------------------------------------
extern "C" void kernel_launch(void* const* d_in, const int* in_sizes, int n_in,
                              void* d_out, int out_size, void* d_ws, size_t ws_size,
                              hipStream_t stream)
{
  const float* x   = (const float*)d_in[0];
  const float* pos = (const float*)d_in[1];
  // d_in[2] = batch (unused: clouds are equal-sized and sorted)
  const float* W1 = (const float*)d_in[3];
  const float* b1 = (const float*)d_in[4];
  const float* W2 = (const float*)d_in[5];
  const float* b2 = (const float*)d_in[6];
  const float* W3 = (const float*)d_in[7];
  const float* b3 = (const float*)d_in[8];
  const float* Wg = (const float*)d_in[9];
  const float* bg = (const float*)d_in[10];

  // Output layout: x_out [8192*128] | pos_out [8192*3] | batch_out [8192] (int bits)
  float* out       = (float*)d_out;
  float* out_x     = out;
  float* out_pos   = out + (size_t)NQ_ * 128;
  int*   out_batch = (int*)(out + (size_t)NQ_ * 128 + (size_t)NQ_ * 3);

  // Workspace: qpos [8192*3 f32] | nidx [8192*32 i32] | vmask [8192 u32] | W1pad [68*64 f32]
  float*    qpos  = (float*)d_ws;
  int*      nidx  = (int*)(qpos + (size_t)NQ_ * 3);
  unsigned* vmask = (unsigned*)(nidx + (size_t)NQ_ * K_);
  float*    W1p   = (float*)(vmask + NQ_);

  pad_w1_kernel<<<(68 * 64 + 255) / 256, 256, 0, stream>>>(W1, W1p);
  fps_kernel<<<B_, 512, 0, stream>>>(pos, qpos, out_pos, out_batch);
  knn_kernel<<<NQ_ / KNN_T, KNN_T, 0, stream>>>(pos, qpos, nidx, vmask);
  mlp_kernel<<<NQ_, 32, 0, stream>>>(x, pos, qpos, nidx, vmask,
                                     W1p, b1, W2, b2, W3, b3, Wg, bg, out_x);
}